// Rayleigh_Bernard_Model_11407433138748
// MI455X (gfx1250) — compile-verified
//
#include <hip/hip_runtime.h>
#include <stdint.h>

#ifndef __has_builtin
#define __has_builtin(x) 0
#endif

#define NXg 256
#define NYg 256
#define NZg 128

typedef int v4i __attribute__((vector_size(16)));
typedef __attribute__((address_space(1))) v4i gas_v4i;  // global, 16B vector
typedef __attribute__((address_space(3))) v4i las_v4i;  // LDS, 16B vector
typedef __attribute__((address_space(1))) int gas_int;  // global
typedef __attribute__((address_space(3))) int las_int;  // LDS

// CDNA5 async global->LDS copy, 16B per lane (tracked by ASYNCcnt).
__device__ __forceinline__ void async_g2l_b128(const float* g, float* l) {
#if __has_builtin(__builtin_amdgcn_global_load_async_to_lds_b128)
  __builtin_amdgcn_global_load_async_to_lds_b128((gas_v4i*)g, (las_v4i*)l, 0, 0);
#elif __has_builtin(__builtin_amdgcn_global_load_async_to_lds_b32)
  // INST_OFFSET applies to BOTH global and LDS addresses (ISA 15.18.3 / CDNA5 ISA)
  __builtin_amdgcn_global_load_async_to_lds_b32((gas_int*)g, (las_int*)l, 0, 0);
  __builtin_amdgcn_global_load_async_to_lds_b32((gas_int*)g, (las_int*)l, 4, 0);
  __builtin_amdgcn_global_load_async_to_lds_b32((gas_int*)g, (las_int*)l, 8, 0);
  __builtin_amdgcn_global_load_async_to_lds_b32((gas_int*)g, (las_int*)l, 12, 0);
#else
  asm volatile("global_load_async_to_lds_b128 %0, %1, off"
               :: "v"((unsigned)(uintptr_t)l), "v"((unsigned long long)(uintptr_t)g)
               : "memory");
#endif
}

__device__ __forceinline__ void wait_asynccnt0() {
#if __has_builtin(__builtin_amdgcn_s_wait_asynccnt)
  __builtin_amdgcn_s_wait_asynccnt(0);
#else
  asm volatile("s_wait_asynccnt 0" ::: "memory");
#endif
}

__device__ __forceinline__ float elem(const float4& v, int k) {
  switch (k & 3) { case 0: return v.x; case 1: return v.y; case 2: return v.z; default: return v.w; }
}
__device__ __forceinline__ void setel(float4& v, int k, float s) {
  switch (k & 3) { case 0: v.x = s; break; case 1: v.y = s; break; case 2: v.z = s; break; default: v.w = s; break; }
}
__device__ __forceinline__ float4 ld4(const float* p) { return *reinterpret_cast<const float4*>(p); }
__device__ __forceinline__ void   st4(float* p, float4 v) { *reinterpret_cast<float4*>(p) = v; }

__global__ __launch_bounds__(128) void rb_rhs_kernel(const float* __restrict__ st,
                                                     float* __restrict__ out) {
  constexpr float CPc = 1004.0f, CVc = 717.0f, MUc = 1.8e-5f, KTHc = 0.025f, Gc = 9.8f;
  constexpr float RGASc = CPc - CVc;  // 287
  constexpr float DXc = 1.0f / NXg, DYc = 1.0f / NYg, DZc = 1.0f / (NZg - 1);
  constexpr float I2DX = 0.5f / DXc, I2DY = 0.5f / DYc, I2DZ = 0.5f / DZc;
  constexpr float IDX2 = 1.0f / (DXc * DXc), IDY2 = 1.0f / (DYc * DYc), IDZ2 = 1.0f / (DZc * DZc);
  constexpr float ICV = 1.0f / CVc;

  const int lane = threadIdx.x & 31;   // lane in wave32
  const int wv   = threadIdx.x >> 5;   // wave id: selects y column
  const int z0   = lane << 2;          // 4 z-points per lane: z0..z0+3
  const int x    = blockIdx.x;
  const int y    = (blockIdx.y << 2) + wv;

  __shared__ __align__(16) float s[4][7][NZg];  // per-wave: u,v,w,rou,T,c,p columns
  float (*col)[NZg] = s[wv];

  const size_t fs = (size_t)NXg * NYg * NZg;
  const size_t cbase = ((size_t)x * NYg + y) * NZg;
  const size_t ic = cbase + (size_t)z0;

  // ---- Stage the 6 center z-columns into LDS via async global->LDS (B128/lane) ----
#pragma unroll
  for (int f = 0; f < 6; ++f)
    async_g2l_b128(st + (size_t)f * fs + ic, &col[f][z0]);
  wait_asynccnt0();
  __syncthreads();

  // Center packs (4 z-points each) from LDS
  const float4 u_c = ld4(&col[0][z0]);
  const float4 v_c = ld4(&col[1][z0]);
  const float4 w_c = ld4(&col[2][z0]);
  const float4 r_c = ld4(&col[3][z0]);
  const float4 T_c = ld4(&col[4][z0]);
  const float4 c_c = ld4(&col[5][z0]);

  // Pressure pack -> LDS so neighbor lanes can read its z-halo
  float4 p_c;
  p_c.x = RGASc * r_c.x * T_c.x;
  p_c.y = RGASc * r_c.y * T_c.y;
  p_c.z = RGASc * r_c.z * T_c.z;
  p_c.w = RGASc * r_c.w * T_c.w;
  st4(&col[6][z0], p_c);
  __syncthreads();

  // One element below / above this lane's pack (clamped; unused at walls)
  const int zb = z0 ? z0 - 1 : 0;
  const int za = (z0 < NZg - 4) ? z0 + 4 : NZg - 1;
  const float u_b = col[0][zb], u_a = col[0][za];
  const float v_b = col[1][zb], v_a = col[1][za];
  const float w_b = col[2][zb], w_a = col[2][za];
  const float T_b = col[4][zb], T_a = col[4][za];
  const float s_b = col[5][zb], s_a = col[5][za];   // scalar c field
  const float p_b = col[6][zb], p_a = col[6][za];

  // ---- Halo columns (x+-1, y+-1, periodic), B128 loads to registers ----
  const int xp = (x + 1) & (NXg - 1), xm = (x - 1) & (NXg - 1);
  const int yp = (y + 1) & (NYg - 1), ym = (y - 1) & (NYg - 1);
  const size_t ixp = ((size_t)xp * NYg + y) * NZg + (size_t)z0;
  const size_t ixm = ((size_t)xm * NYg + y) * NZg + (size_t)z0;
  const size_t iyp = ((size_t)x * NYg + yp) * NZg + (size_t)z0;
  const size_t iym = ((size_t)x * NYg + ym) * NZg + (size_t)z0;

  const float4 u_xp = ld4(st + 0 * fs + ixp), u_xm = ld4(st + 0 * fs + ixm);
  const float4 u_yp = ld4(st + 0 * fs + iyp), u_ym = ld4(st + 0 * fs + iym);
  const float4 v_xp = ld4(st + 1 * fs + ixp), v_xm = ld4(st + 1 * fs + ixm);
  const float4 v_yp = ld4(st + 1 * fs + iyp), v_ym = ld4(st + 1 * fs + iym);
  const float4 w_xp = ld4(st + 2 * fs + ixp), w_xm = ld4(st + 2 * fs + ixm);
  const float4 w_yp = ld4(st + 2 * fs + iyp), w_ym = ld4(st + 2 * fs + iym);
  const float4 r_xp = ld4(st + 3 * fs + ixp), r_xm = ld4(st + 3 * fs + ixm);
  const float4 r_yp = ld4(st + 3 * fs + iyp), r_ym = ld4(st + 3 * fs + iym);
  const float4 T_xp = ld4(st + 4 * fs + ixp), T_xm = ld4(st + 4 * fs + ixm);
  const float4 T_yp = ld4(st + 4 * fs + iyp), T_ym = ld4(st + 4 * fs + iym);
  const float4 c_xp = ld4(st + 5 * fs + ixp), c_xm = ld4(st + 5 * fs + ixm);
  const float4 c_yp = ld4(st + 5 * fs + iyp), c_ym = ld4(st + 5 * fs + iym);

  float4 Ru, Rv, Rw, Rr, RT, Rs;

#pragma unroll
  for (int k = 0; k < 4; ++k) {
    // wall conditions: statically gated on k so they only exist for k==0 / k==3
    const bool zlo = (k == 0) && (z0 == 0);
    const bool zhi = (k == 3) && (z0 == NZg - 4);

    // z first/second derivative from the pack + one halo element each side
    auto d1d2z = [&](const float4& f4, float fb, float fa, float& d1, float& d2) {
      const float fm = (k == 0) ? fb : elem(f4, k - 1);
      const float fp = (k == 3) ? fa : elem(f4, k + 1);
      const float fk = elem(f4, k);
      float c1 = fp - fm;
      float c2 = fp - 2.0f * fk + fm;
      if (zlo) {
        c1 = -3.0f * elem(f4, 0) + 4.0f * elem(f4, 1) - elem(f4, 2);
        c2 =  2.0f * elem(f4, 0) - 5.0f * elem(f4, 1) + 4.0f * elem(f4, 2) - elem(f4, 3);
      } else if (zhi) {
        c1 =  3.0f * elem(f4, 3) - 4.0f * elem(f4, 2) + elem(f4, 1);
        c2 =  2.0f * elem(f4, 3) - 5.0f * elem(f4, 2) + 4.0f * elem(f4, 1) - elem(f4, 0);
      }
      d1 = c1 * I2DZ;
      d2 = c2 * IDZ2;
    };
    auto d1z_only = [&](const float4& f4, float fb, float fa) -> float {
      const float fm = (k == 0) ? fb : elem(f4, k - 1);
      const float fp = (k == 3) ? fa : elem(f4, k + 1);
      float c1 = fp - fm;
      if (zlo)      c1 = -3.0f * elem(f4, 0) + 4.0f * elem(f4, 1) - elem(f4, 2);
      else if (zhi) c1 =  3.0f * elem(f4, 3) - 4.0f * elem(f4, 2) + elem(f4, 1);
      return c1 * I2DZ;
    };

    const float uk = elem(u_c, k), vk = elem(v_c, k), wk = elem(w_c, k);
    const float rk = elem(r_c, k), Tk = elem(T_c, k);
    const float invr = 1.0f / rk;

    float du1, du2, dv1, dv2, dw1, dw2, dT1, dT2;
    d1d2z(u_c, u_b, u_a, du1, du2);
    d1d2z(v_c, v_b, v_a, dv1, dv2);
    d1d2z(w_c, w_b, w_a, dw1, dw2);
    d1d2z(T_c, T_b, T_a, dT1, dT2);
    const float dc1 = d1z_only(c_c, s_b, s_a);
    const float dp1 = d1z_only(p_c, p_b, p_a);

    const float uxpk = elem(u_xp, k), uxmk = elem(u_xm, k), uypk = elem(u_yp, k), uymk = elem(u_ym, k);
    const float vxpk = elem(v_xp, k), vxmk = elem(v_xm, k), vypk = elem(v_yp, k), vymk = elem(v_ym, k);
    const float wxpk = elem(w_xp, k), wxmk = elem(w_xm, k), wypk = elem(w_yp, k), wymk = elem(w_ym, k);
    const float rxpk = elem(r_xp, k), rxmk = elem(r_xm, k), rypk = elem(r_yp, k), rymk = elem(r_ym, k);
    const float Txpk = elem(T_xp, k), Txmk = elem(T_xm, k), Typk = elem(T_yp, k), Tymk = elem(T_ym, k);
    const float cxpk = elem(c_xp, k), cxmk = elem(c_xm, k), cypk = elem(c_yp, k), cymk = elem(c_ym, k);

    const float pxpk = RGASc * rxpk * Txpk, pxmk = RGASc * rxmk * Txmk;
    const float pypk = RGASc * rypk * Typk, pymk = RGASc * rymk * Tymk;

    // du
    const float lap_u = (uxpk + uxmk - 2.0f * uk) * IDX2 + (uypk + uymk - 2.0f * uk) * IDY2 + du2;
    const float adv_u = uk * ((uxpk - uxmk) * I2DX) + vk * ((uypk - uymk) * I2DY) + wk * du1;
    float duv = (-(pxpk - pxmk) * I2DX + MUc * lap_u) * invr - adv_u;

    // dv
    const float lap_v = (vxpk + vxmk - 2.0f * vk) * IDX2 + (vypk + vymk - 2.0f * vk) * IDY2 + dv2;
    const float adv_v = uk * ((vxpk - vxmk) * I2DX) + vk * ((vypk - vymk) * I2DY) + wk * dv1;
    float dvv = (-(pypk - pymk) * I2DY + MUc * lap_v) * invr - adv_v;

    // dw
    const float lap_w = (wxpk + wxmk - 2.0f * wk) * IDX2 + (wypk + wymk - 2.0f * wk) * IDY2 + dw2;
    const float adv_w = uk * ((wxpk - wxmk) * I2DX) + vk * ((wypk - wymk) * I2DY) + wk * dw1;
    float dwv = (-Gc * rk - dp1 + MUc * lap_w) * invr - adv_w;

    // drou (reference only keeps the x flux term)
    const float drv = -((rxpk * uxpk) - (rxmk * uxmk)) * I2DX;

    // dT
    const float lap_T = (Txpk + Txmk - 2.0f * Tk) * IDX2 + (Typk + Tymk - 2.0f * Tk) * IDY2 + dT2;
    const float adv_T = uk * ((Txpk - Txmk) * I2DX) + vk * ((Typk - Tymk) * I2DY) + wk * dT1;
    float dTv = KTHc * lap_T * invr * ICV - adv_T;

    // dc
    const float adv_c = uk * ((cxpk - cxmk) * I2DX) + vk * ((cypk - cymk) * I2DY) + wk * dc1;
    const float dcv = -adv_c;

    if (zlo || zhi) { duv = 0.0f; dvv = 0.0f; dwv = 0.0f; dTv = 0.0f; }

    setel(Ru, k, duv);
    setel(Rv, k, dvv);
    setel(Rw, k, dwv);
    setel(Rr, k, drv);
    setel(RT, k, dTv);
    setel(Rs, k, dcv);
  }

  st4(out + 0 * fs + ic, Ru);
  st4(out + 1 * fs + ic, Rv);
  st4(out + 2 * fs + ic, Rw);
  st4(out + 3 * fs + ic, Rr);
  st4(out + 4 * fs + ic, RT);
  st4(out + 5 * fs + ic, Rs);
}

extern "C" void kernel_launch(void* const* d_in, const int* in_sizes, int n_in,
                              void* d_out, int out_size, void* d_ws, size_t ws_size,
                              hipStream_t stream) {
  const float* st = (const float*)d_in[0];  // state [6,256,256,128] f32
  // d_in[1] is t (unused by the RHS)
  float* out = (float*)d_out;               // [6,256,256,128] f32
  (void)in_sizes; (void)n_in; (void)d_ws; (void)ws_size; (void)out_size;

  dim3 grid(NXg, NYg / 4, 1);
  dim3 block(128, 1, 1);  // 4 wave32; each wave owns one full z-column (float4/lane)
  rb_rhs_kernel<<<grid, block, 0, stream>>>(st, out);
}